// get_model_6416681140703
// MI455X (gfx1250) — compile-verified
//
#include <hip/hip_runtime.h>
#include <hip/hip_bf16.h>

typedef float v2f __attribute__((ext_vector_type(2)));
typedef float v8f __attribute__((ext_vector_type(8)));

#define NB    4
#define NPTS  4096
#define KNN_K 20
#define NSLOPE 0.2f
#define EPSV  1e-6f
#define BNEPS 1e-5f

#if defined(__has_builtin)
#if __has_builtin(__builtin_amdgcn_sched_barrier)
#define SCHED_FENCE() __builtin_amdgcn_sched_barrier(0)
#endif
#endif
#ifndef SCHED_FENCE
#define SCHED_FENCE()
#endif

// ---------------------------------------------------------------------------
// Generic f32 WMMA SGEMM:  C[m,col] = sum_k A[m,k]*B[k,col] (+ bias[m])
// A: M x Kd row-major (lda), shared across batches.
// B: per-batch at Bm + b*b_bstride, row-major Kd x Ncols (ldb).
// C: per-batch at Cm + b*c_bstride, element (m,col) at m*c_rs + col*c_cs.
// REQUIRES: Ncols % 256 == 0 (true for every call site here).
// A-rows >= M are clamped to row M-1 (WMMA row m of D depends only on row m
// of A, and rows >= M are never stored).
// Each wave computes a 16x64 tile: 4 accumulators, 4 WMMAs per k-step.
// Software-pipelined with a sched_barrier so the loads of step i+1 stay
// ahead of the WMMAs of step i -> waitcnt drains only the older load group.
// ---------------------------------------------------------------------------
__global__ __launch_bounds__(128) void sgemm_wmma_kernel(
    const float* __restrict__ A, int lda,
    const float* __restrict__ Bm, long b_bstride, int ldb,
    float* __restrict__ Cm, long c_bstride, long c_rs, int c_cs,
    const float* __restrict__ bias, int bias_bstride,
    int M, int Kd, int Ncols)
{
    const int lane = threadIdx.x & 31;
    const int wave = threadIdx.x >> 5;
    const int b    = blockIdx.z;
    const int m0   = blockIdx.y * 16;
    const int colw = blockIdx.x * 256 + wave * 64;

    const float* __restrict__ Bb = Bm + (long)b * b_bstride;
    float* __restrict__ Cb = Cm + (long)b * c_bstride;

    const int arow = lane & 15;
    const int kb   = (lane >> 4) << 1;          // 0 or 2
    const int bcol = colw + (lane & 15);

    int arow_c = m0 + arow;
    if (arow_c > M - 1) arow_c = M - 1;         // clamp: garbage rows never stored
    const float* __restrict__ Arow = A + (long)arow_c * lda;
    const float* __restrict__ Ap = Arow + kb;
    const float* __restrict__ Bp = Bb + (long)kb * ldb + bcol;
    const long ldb4 = 4L * ldb;

    v8f acc0 = {0.f,0.f,0.f,0.f,0.f,0.f,0.f,0.f};
    v8f acc1 = acc0, acc2 = acc0, acc3 = acc0;

    const int Kmain = Kd & ~3;
    if (Kmain > 0) {
        // prologue: load fragments for ko = 0
        v2f a, b0, b1, b2, b3;
        a.x  = Ap[0];        a.y  = Ap[1];
        b0.x = Bp[0];        b0.y = Bp[ldb];
        b1.x = Bp[16];       b1.y = Bp[ldb + 16];
        b2.x = Bp[32];       b2.y = Bp[ldb + 32];
        b3.x = Bp[48];       b3.y = Bp[ldb + 48];
        for (int ko = 4; ko < Kmain; ko += 4) {
            Ap += 4;
            Bp += ldb4;
            // issue next-step loads; fence keeps them ahead of this step's WMMAs
            v2f an, b0n, b1n, b2n, b3n;
            an.x  = Ap[0];   an.y  = Ap[1];
            b0n.x = Bp[0];   b0n.y = Bp[ldb];
            b1n.x = Bp[16];  b1n.y = Bp[ldb + 16];
            b2n.x = Bp[32];  b2n.y = Bp[ldb + 32];
            b3n.x = Bp[48];  b3n.y = Bp[ldb + 48];
            SCHED_FENCE();
            acc0 = __builtin_amdgcn_wmma_f32_16x16x4_f32(false, a, false, b0, (short)0, acc0, false, false);
            acc1 = __builtin_amdgcn_wmma_f32_16x16x4_f32(false, a, false, b1, (short)0, acc1, false, false);
            acc2 = __builtin_amdgcn_wmma_f32_16x16x4_f32(false, a, false, b2, (short)0, acc2, false, false);
            acc3 = __builtin_amdgcn_wmma_f32_16x16x4_f32(false, a, false, b3, (short)0, acc3, false, false);
            SCHED_FENCE();
            a = an; b0 = b0n; b1 = b1n; b2 = b2n; b3 = b3n;
        }
        acc0 = __builtin_amdgcn_wmma_f32_16x16x4_f32(false, a, false, b0, (short)0, acc0, false, false);
        acc1 = __builtin_amdgcn_wmma_f32_16x16x4_f32(false, a, false, b1, (short)0, acc1, false, false);
        acc2 = __builtin_amdgcn_wmma_f32_16x16x4_f32(false, a, false, b2, (short)0, acc2, false, false);
        acc3 = __builtin_amdgcn_wmma_f32_16x16x4_f32(false, a, false, b3, (short)0, acc3, false, false);
    }
    if (Kmain < Kd) {
        // K-tail: clamp indices in-bounds (unconditional loads) + zero-select.
        const int k0 = Kmain + kb, k1 = k0 + 1;
        const bool k0ok = k0 < Kd, k1ok = k1 < Kd;
        const int k0c = k0ok ? k0 : 0;
        const int k1c = k1ok ? k1 : 0;
        const float* __restrict__ B0 = Bb + (long)k0c * ldb + bcol;
        const float* __restrict__ B1 = Bb + (long)k1c * ldb + bcol;
        v2f a, b0, b1, b2, b3;
        const float ax = Arow[k0c], ay = Arow[k1c];
        a.x  = k0ok ? ax : 0.f;
        a.y  = k1ok ? ay : 0.f;
        const float u0 = B0[0],  v0 = B1[0];
        const float u1 = B0[16], v1 = B1[16];
        const float u2 = B0[32], v2 = B1[32];
        const float u3 = B0[48], v3 = B1[48];
        b0.x = k0ok ? u0 : 0.f;  b0.y = k1ok ? v0 : 0.f;
        b1.x = k0ok ? u1 : 0.f;  b1.y = k1ok ? v1 : 0.f;
        b2.x = k0ok ? u2 : 0.f;  b2.y = k1ok ? v2 : 0.f;
        b3.x = k0ok ? u3 : 0.f;  b3.y = k1ok ? v3 : 0.f;
        acc0 = __builtin_amdgcn_wmma_f32_16x16x4_f32(false, a, false, b0, (short)0, acc0, false, false);
        acc1 = __builtin_amdgcn_wmma_f32_16x16x4_f32(false, a, false, b1, (short)0, acc1, false, false);
        acc2 = __builtin_amdgcn_wmma_f32_16x16x4_f32(false, a, false, b2, (short)0, acc2, false, false);
        acc3 = __builtin_amdgcn_wmma_f32_16x16x4_f32(false, a, false, b3, (short)0, acc3, false, false);
    }

    const int mh = (lane >> 4) << 3;
    const int bcol1 = bcol + 16, bcol2 = bcol + 32, bcol3 = bcol + 48;
#pragma unroll
    for (int v = 0; v < 8; ++v) {
        const int m = m0 + mh + v;
        if (m < M) {
            const float bv = bias ? bias[(long)b * bias_bstride + m] : 0.f;
            const long mo = (long)m * c_rs;
            if (bcol  < Ncols) Cb[mo + (long)bcol  * c_cs] = acc0[v] + bv;
            if (bcol1 < Ncols) Cb[mo + (long)bcol1 * c_cs] = acc1[v] + bv;
            if (bcol2 < Ncols) Cb[mo + (long)bcol2 * c_cs] = acc2[v] + bv;
            if (bcol3 < Ncols) Cb[mo + (long)bcol3 * c_cs] = acc3[v] + bv;
        }
    }
}

// ---------------------------------------------------------------------------
// KNN: one wave per query point n. x rows: d x N at x + b*xbs, row r at r*N.
// value = -||x_n - x_m||^2, keep top-20 indices (self included, dist 0).
// ---------------------------------------------------------------------------
__global__ __launch_bounds__(128) void knn_kernel(
    const float* __restrict__ x, int d, long xbs, int* __restrict__ idxout)
{
    const int N = NPTS;
    const int wave = threadIdx.x >> 5;
    const int lane = threadIdx.x & 31;
    const int n = blockIdx.x * 4 + wave;
    const int b = blockIdx.y;
    const float* __restrict__ xb = x + (long)b * xbs;

    __shared__ float q[4][64];
    __shared__ float sval[4][32 * KNN_K];
    __shared__ int   sidx[4][32 * KNN_K];

    for (int c = lane; c < d; c += 32) q[wave][c] = xb[(long)c * N + n];
    __syncthreads();

    float val[KNN_K]; int ind[KNN_K];
#pragma unroll
    for (int j = 0; j < KNN_K; ++j) { val[j] = -3.4e38f; ind[j] = 0; }

    for (int m = lane; m < N; m += 32) {
        float acc = 0.f;
        for (int c = 0; c < d; ++c) {
            const float diff = q[wave][c] - xb[(long)c * N + m];
            acc = fmaf(diff, diff, acc);
        }
        const float v = -acc;
        if (v > val[KNN_K - 1]) {
            val[KNN_K - 1] = v; ind[KNN_K - 1] = m;
#pragma unroll
            for (int j = KNN_K - 1; j > 0; --j) {
                if (val[j] > val[j - 1]) {
                    const float tv = val[j]; val[j] = val[j - 1]; val[j - 1] = tv;
                    const int ti = ind[j]; ind[j] = ind[j - 1]; ind[j - 1] = ti;
                }
            }
        }
    }

#pragma unroll
    for (int j = 0; j < KNN_K; ++j) {
        sval[wave][lane * KNN_K + j] = val[j];
        sidx[wave][lane * KNN_K + j] = ind[j];
    }
    __syncthreads();

    int ptr = 0;
    int* __restrict__ outp = idxout + ((long)b * N + n) * KNN_K;
    for (int r = 0; r < KNN_K; ++r) {
        float bv = sval[wave][lane * KNN_K + ptr];
        int bl = lane;
#pragma unroll
        for (int off = 16; off > 0; off >>= 1) {
            const float ov = __shfl_xor(bv, off, 32);
            const int   ol = __shfl_xor(bl, off, 32);
            if (ov > bv || (ov == bv && ol < bl)) { bv = ov; bl = ol; }
        }
        const int pw = __shfl(ptr, bl, 32);
        if (lane == 0) outp[r] = sidx[wave][bl * KNN_K + pw];
        if (lane == bl) ++ptr;
    }
}

// ---------------------------------------------------------------------------
// Graph feature: out (B, 2C, 3, N, K); ch<C: x[.,idx]-x[.,n]; ch>=C: x[.,n]
// ---------------------------------------------------------------------------
__global__ void graph_feat_kernel(const float* __restrict__ x, long xbs,
                                  const int* __restrict__ idx,
                                  float* __restrict__ out, int Cc)
{
    const int N = NPTS;
    const long tot = (long)NB * Cc * 3 * N * KNN_K;
    const long tid = (long)blockIdx.x * blockDim.x + threadIdx.x;
    if (tid >= tot) return;
    const int k = (int)(tid % KNN_K);
    long t1 = tid / KNN_K;
    const int n = (int)(t1 % N); t1 /= N;
    const int t = (int)(t1 % 3); t1 /= 3;
    const int c = (int)(t1 % Cc);
    const int b = (int)(t1 / Cc);

    const float* __restrict__ xb = x + (long)b * xbs;
    const int j = idx[((long)b * N + n) * KNN_K + k];
    const float xc = xb[(long)(c * 3 + t) * N + n];
    const float xj = xb[(long)(c * 3 + t) * N + j];
    const long nk = (long)N * KNN_K;
    float* __restrict__ ob = out + (long)b * (2 * Cc) * 3 * nk;
    const long col = (long)n * KNN_K + k;
    ob[((long)(c) * 3 + t) * nk + col]       = xj - xc;
    ob[((long)(Cc + c) * 3 + t) * nk + col]  = xc;
}

// ---------------------------------------------------------------------------
// VN-BN pass 1: norms + per-channel (sum, sumsq) via block reduce + atomics.
// grid: (ceil(M/1024), Cout, B), block 256.
// ---------------------------------------------------------------------------
__global__ __launch_bounds__(256) void vn_norm_stats_kernel(
    const float* __restrict__ p, long pbs, float* __restrict__ nbuf,
    float* __restrict__ stats, int Cout, int Mlen)
{
    const int c = blockIdx.y, b = blockIdx.z;
    const float* __restrict__ pb = p + (long)b * pbs + (long)c * 3 * Mlen;
    float* __restrict__ nb = nbuf + ((long)b * Cout + c) * Mlen;
    float s1 = 0.f, s2 = 0.f;
    const int base = blockIdx.x * 1024;
#pragma unroll
    for (int it = 0; it < 4; ++it) {
        const int m = base + it * 256 + threadIdx.x;
        if (m < Mlen) {
            const float a0 = pb[m];
            const float a1 = pb[(long)Mlen + m];
            const float a2 = pb[2L * Mlen + m];
            const float nv = sqrtf(a0 * a0 + a1 * a1 + a2 * a2) + EPSV;
            nb[m] = nv;
            s1 += nv; s2 += nv * nv;
        }
    }
    __shared__ float sh1[256], sh2[256];
    sh1[threadIdx.x] = s1; sh2[threadIdx.x] = s2;
    __syncthreads();
    for (int off = 128; off > 0; off >>= 1) {
        if ((int)threadIdx.x < off) {
            sh1[threadIdx.x] += sh1[threadIdx.x + off];
            sh2[threadIdx.x] += sh2[threadIdx.x + off];
        }
        __syncthreads();
    }
    if (threadIdx.x == 0) {
        atomicAdd(&stats[2 * c],     sh1[0]);
        atomicAdd(&stats[2 * c + 1], sh2[0]);
    }
}

// ---------------------------------------------------------------------------
// VN-BN pass 2 + leaky combine. Cd==1 -> shared direction channel.
// ---------------------------------------------------------------------------
__global__ void vn_finalize_kernel(
    const float* __restrict__ p, long pbs,
    const float* __restrict__ dmat, long dbs, int Cd,
    const float* __restrict__ nbuf, const float* __restrict__ stats,
    const float* __restrict__ gamma, const float* __restrict__ beta,
    float invcount, float* __restrict__ out, long obs, int Cout, int Mlen)
{
    const long tot = (long)NB * Cout * Mlen;
    const long tid = (long)blockIdx.x * blockDim.x + threadIdx.x;
    if (tid >= tot) return;
    const int m = (int)(tid % Mlen);
    long t1 = tid / Mlen;
    const int c = (int)(t1 % Cout);
    const int b = (int)(t1 / Cout);

    const float* __restrict__ pb = p + (long)b * pbs + (long)c * 3 * Mlen;
    const int cd = (Cd == 1) ? 0 : c;
    const float* __restrict__ db = dmat + (long)b * dbs + (long)cd * 3 * Mlen;

    const float nv = nbuf[((long)b * Cout + c) * Mlen + m];
    const float s1 = stats[2 * c], s2 = stats[2 * c + 1];
    const float mean = s1 * invcount;
    const float var = s2 * invcount - mean * mean;
    const float nbv = (nv - mean) * rsqrtf(var + BNEPS) * gamma[c] + beta[c];
    const float sc = nbv / nv;
    const float p0 = pb[m] * sc, p1 = pb[(long)Mlen + m] * sc, p2 = pb[2L * Mlen + m] * sc;
    const float d0 = db[m],      d1 = db[(long)Mlen + m],      d2 = db[2L * Mlen + m];
    const float dot = p0 * d0 + p1 * d1 + p2 * d2;
    const float dsq = d0 * d0 + d1 * d1 + d2 * d2;
    const float f = (dot >= 0.f) ? 0.f : dot / (dsq + EPSV);
    float* __restrict__ ob = out + (long)b * obs + (long)c * 3 * Mlen;
    ob[m]              = NSLOPE * p0 + (1.f - NSLOPE) * (p0 - f * d0);
    ob[(long)Mlen + m] = NSLOPE * p1 + (1.f - NSLOPE) * (p1 - f * d1);
    ob[2L * Mlen + m]  = NSLOPE * p2 + (1.f - NSLOPE) * (p2 - f * d2);
}

// mean over K axis: h (B,21,3,N,K) -> x123 channel slice [coff, coff+21)
__global__ void mean_k_kernel(const float* __restrict__ h, float* __restrict__ x123, int coff)
{
    const int N = NPTS, Cc = 21;
    const long tot = (long)NB * Cc * 3 * N;
    const long tid = (long)blockIdx.x * blockDim.x + threadIdx.x;
    if (tid >= tot) return;
    const int n = (int)(tid % N);
    long t1 = tid / N;
    const int t = (int)(t1 % 3); t1 /= 3;
    const int c = (int)(t1 % Cc);
    const int b = (int)(t1 / Cc);
    const float* __restrict__ hb =
        h + ((long)((b * Cc + c) * 3 + t) * N + n) * KNN_K;
    float s = 0.f;
#pragma unroll
    for (int k = 0; k < KNN_K; ++k) s += hb[k];
    x123[((long)(b * 63 + coff + c) * 3 + t) * N + n] = s * (1.f / (float)KNN_K);
}

// mean over N (broadcast) for h6cat channels [341,682). grid (1023, B)
__global__ __launch_bounds__(256) void hmean_cat_kernel(float* __restrict__ h6)
{
    const int N = NPTS;
    const int rc = blockIdx.x;           // c*3 + t, rc in [0, 1023)
    const int b = blockIdx.y;
    float* __restrict__ rowin  = h6 + ((long)b * 2046 + rc) * N;
    float* __restrict__ rowout = rowin + 1023L * N;
    float s = 0.f;
    for (int n = threadIdx.x; n < N; n += 256) s += rowin[n];
    __shared__ float sh[256];
    sh[threadIdx.x] = s;
    __syncthreads();
    for (int off = 128; off > 0; off >>= 1) {
        if ((int)threadIdx.x < off) sh[threadIdx.x] += sh[threadIdx.x + off];
        __syncthreads();
    }
    const float mean = sh[0] * (1.f / (float)N);
    for (int n = threadIdx.x; n < N; n += 256) rowout[n] = mean;
}

// fused hs = h6 (.,3,n) x z0t(3,3,n) and max over n. grid (682, B)
__global__ __launch_bounds__(256) void hsmax_kernel(
    const float* __restrict__ h6, const float* __restrict__ z0, float* __restrict__ hmax)
{
    const int N = NPTS;
    const int i = blockIdx.x, b = blockIdx.y;
    const float* __restrict__ hb = h6 + ((long)b * 682 + i) * 3L * N;
    const float* __restrict__ zb = z0 + (long)b * 9L * N;
    float mx0 = -3.4e38f, mx1 = -3.4e38f, mx2 = -3.4e38f;
    for (int n = threadIdx.x; n < N; n += 256) {
        const float h0 = hb[n], h1 = hb[(long)N + n], h2 = hb[2L * N + n];
        const float v0 = h0 * zb[n]          + h1 * zb[(long)N + n]  + h2 * zb[2L * N + n];
        const float v1 = h0 * zb[3L * N + n] + h1 * zb[4L * N + n]   + h2 * zb[5L * N + n];
        const float v2 = h0 * zb[6L * N + n] + h1 * zb[7L * N + n]   + h2 * zb[8L * N + n];
        mx0 = fmaxf(mx0, v0); mx1 = fmaxf(mx1, v1); mx2 = fmaxf(mx2, v2);
    }
    __shared__ float s0[256], s1[256], s2[256];
    s0[threadIdx.x] = mx0; s1[threadIdx.x] = mx1; s2[threadIdx.x] = mx2;
    __syncthreads();
    for (int off = 128; off > 0; off >>= 1) {
        if ((int)threadIdx.x < off) {
            s0[threadIdx.x] = fmaxf(s0[threadIdx.x], s0[threadIdx.x + off]);
            s1[threadIdx.x] = fmaxf(s1[threadIdx.x], s1[threadIdx.x + off]);
            s2[threadIdx.x] = fmaxf(s2[threadIdx.x], s2[threadIdx.x + off]);
        }
        __syncthreads();
    }
    if (threadIdx.x == 0) {
        hmax[(long)b * 2046 + i * 3 + 0] = s0[0];
        hmax[(long)b * 2046 + i * 3 + 1] = s1[0];
        hmax[(long)b * 2046 + i * 3 + 2] = s2[0];
    }
}

// x123p[b, i*3+k, n] = sum_j x123[b,i,j,n] * z0[b,k,j,n]
__global__ void x123p_kernel(const float* __restrict__ x123,
                             const float* __restrict__ z0, float* __restrict__ out)
{
    const int N = NPTS;
    const long tot = (long)NB * 63 * N;
    const long tid = (long)blockIdx.x * blockDim.x + threadIdx.x;
    if (tid >= tot) return;
    const int n = (int)(tid % N);
    long t1 = tid / N;
    const int i = (int)(t1 % 63);
    const int b = (int)(t1 / 63);
    const float* __restrict__ xb = x123 + ((long)b * 63 + i) * 3L * N;
    const float* __restrict__ zb = z0 + (long)b * 9L * N;
    const float h0 = xb[n], h1 = xb[(long)N + n], h2 = xb[2L * N + n];
    float* __restrict__ ob = out + ((long)b * 189 + i * 3) * N + n;
    ob[0]        = h0 * zb[n]          + h1 * zb[(long)N + n] + h2 * zb[2L * N + n];
    ob[(long)N]  = h0 * zb[3L * N + n] + h1 * zb[4L * N + n]  + h2 * zb[5L * N + n];
    ob[2L * N]   = h0 * zb[6L * N + n] + h1 * zb[7L * N + n]  + h2 * zb[8L * N + n];
}

// bias8[b,o] = sum_{i<2046} w8[o,i] * hmax[b,i]. grid (256, B)
__global__ __launch_bounds__(256) void bias8_kernel(
    const float* __restrict__ w8, const float* __restrict__ hmax, float* __restrict__ bias8)
{
    const int o = blockIdx.x, b = blockIdx.y;
    const float* __restrict__ wrow = w8 + (long)o * 2235;
    const float* __restrict__ hm = hmax + (long)b * 2046;
    float s = 0.f;
    for (int i = threadIdx.x; i < 2046; i += 256) s += wrow[i] * hm[i];
    __shared__ float sh[256];
    sh[threadIdx.x] = s;
    __syncthreads();
    for (int off = 128; off > 0; off >>= 1) {
        if ((int)threadIdx.x < off) sh[threadIdx.x] += sh[threadIdx.x + off];
        __syncthreads();
    }
    if (threadIdx.x == 0) bias8[(long)b * 256 + o] = sh[0];
}

// BN1d stats: x (B, Cc, N), per-channel sum/sumsq. grid (4, Cc, B)
__global__ __launch_bounds__(256) void bn_stats_kernel(
    const float* __restrict__ x, int Cc, float* __restrict__ stats)
{
    const int N = NPTS;
    const int c = blockIdx.y, b = blockIdx.z;
    const float* __restrict__ xb = x + ((long)b * Cc + c) * N;
    float s1 = 0.f, s2 = 0.f;
    const int base = blockIdx.x * 1024;
#pragma unroll
    for (int it = 0; it < 4; ++it) {
        const float v = xb[base + it * 256 + threadIdx.x];
        s1 += v; s2 += v * v;
    }
    __shared__ float sh1[256], sh2[256];
    sh1[threadIdx.x] = s1; sh2[threadIdx.x] = s2;
    __syncthreads();
    for (int off = 128; off > 0; off >>= 1) {
        if ((int)threadIdx.x < off) {
            sh1[threadIdx.x] += sh1[threadIdx.x + off];
            sh2[threadIdx.x] += sh2[threadIdx.x + off];
        }
        __syncthreads();
    }
    if (threadIdx.x == 0) {
        atomicAdd(&stats[2 * c],     sh1[0]);
        atomicAdd(&stats[2 * c + 1], sh2[0]);
    }
}

// BN apply + optional leaky, in-place on (B, Cc, N)
__global__ void bn_apply_kernel(float* __restrict__ x, int Cc,
                                const float* __restrict__ stats,
                                const float* __restrict__ g, const float* __restrict__ bt,
                                float invcount, int do_leaky)
{
    const int N = NPTS;
    const long tot = (long)NB * Cc * N;
    const long tid = (long)blockIdx.x * blockDim.x + threadIdx.x;
    if (tid >= tot) return;
    const int c = (int)((tid / N) % Cc);
    const float mean = stats[2 * c] * invcount;
    const float var = stats[2 * c + 1] * invcount - mean * mean;
    float v = (x[tid] - mean) * rsqrtf(var + BNEPS) * g[c] + bt[c];
    if (do_leaky) v = (v >= 0.f) ? v : NSLOPE * v;
    x[tid] = v;
}

// ---------------------------------------------------------------------------
extern "C" void kernel_launch(void* const* d_in, const int* in_sizes, int n_in,
                              void* d_out, int out_size, void* d_ws, size_t ws_size,
                              hipStream_t stream)
{
    (void)in_sizes; (void)n_in; (void)out_size; (void)ws_size;
    const int B = NB, N = NPTS, C = 21, H = 341, H2 = 170;
    const long NK = (long)N * KNN_K;

    const float* x   = (const float*)d_in[0];
    const float* w1f = (const float*)d_in[1];  const float* w1d = (const float*)d_in[2];
    const float* g1  = (const float*)d_in[3];  const float* b1  = (const float*)d_in[4];
    const float* w2f = (const float*)d_in[5];  const float* w2d = (const float*)d_in[6];
    const float* g2  = (const float*)d_in[7];  const float* b2  = (const float*)d_in[8];
    const float* w3f = (const float*)d_in[9];  const float* w3d = (const float*)d_in[10];
    const float* g3  = (const float*)d_in[11]; const float* b3  = (const float*)d_in[12];
    const float* w4f = (const float*)d_in[13]; const float* w4d = (const float*)d_in[14];
    const float* g4  = (const float*)d_in[15]; const float* b4  = (const float*)d_in[16];
    const float* w5f = (const float*)d_in[17]; const float* w5d = (const float*)d_in[18];
    const float* g5  = (const float*)d_in[19]; const float* b5  = (const float*)d_in[20];
    const float* w6f = (const float*)d_in[21]; const float* w6d = (const float*)d_in[22];
    const float* g6  = (const float*)d_in[23]; const float* b6  = (const float*)d_in[24];
    const float* s1f = (const float*)d_in[25]; const float* s1d = (const float*)d_in[26];
    const float* sg1 = (const float*)d_in[27]; const float* sb1 = (const float*)d_in[28];
    const float* s2f = (const float*)d_in[29]; const float* s2d = (const float*)d_in[30];
    const float* sg2 = (const float*)d_in[31]; const float* sb2 = (const float*)d_in[32];
    const float* slin= (const float*)d_in[33];
    const float* w8  = (const float*)d_in[34]; const float* g8  = (const float*)d_in[35];
    const float* b8  = (const float*)d_in[36];
    const float* w9  = (const float*)d_in[37]; const float* g9  = (const float*)d_in[38];
    const float* b9  = (const float*)d_in[39];
    const float* w10 = (const float*)d_in[40]; const float* g10 = (const float*)d_in[41];
    const float* b10 = (const float*)d_in[42];
    const float* w11 = (const float*)d_in[43];
    float* outp = (float*)d_out;

    char* wsp = (char*)d_ws;
    auto alloc = [&](size_t nelem) -> float* {
        float* p = (float*)wsp;
        wsp += ((nelem * sizeof(float)) + 255) / 256 * 256;
        return p;
    };
    float* gf    = alloc((size_t)B * 42 * 3 * NK);   // also aliases h6cat (B,682,3,N)
    float* pA    = alloc((size_t)B * 21 * 3 * NK);
    float* pB    = alloc((size_t)B * 21 * 3 * NK);
    float* dbuf  = alloc((size_t)B * 21 * 3 * NK);
    float* nbuf  = alloc((size_t)B * 21 * NK);
    float* x123  = alloc((size_t)B * 63 * 3 * N);
    float* z0    = alloc((size_t)B * 9 * N);
    float* xp    = alloc((size_t)B * 189 * N);
    float* hmax  = alloc((size_t)B * 2046);
    float* bias8 = alloc((size_t)B * 256);
    float* h8    = alloc((size_t)B * 256 * N);
    float* h9    = alloc((size_t)B * 256 * N);
    float* h10   = alloc((size_t)B * 128 * N);
    float* stats = alloc((size_t)2 * 341);
    int* idx = (int*)alloc((size_t)B * N * KNN_K);   // int fits in float slot
    float* h6cat = gf;

    auto gemm = [&](const float* A, int lda, const float* Bm, long bbs, int ldb,
                    float* Cm, long cbs, long crs, int ccs,
                    const float* bias, int biasbs, int M, int Kd, int Ncols) {
        dim3 g((Ncols + 255) / 256, (M + 15) / 16, B);
        sgemm_wmma_kernel<<<g, 128, 0, stream>>>(A, lda, Bm, bbs, ldb, Cm, cbs, crs, ccs,
                                                 bias, biasbs, M, Kd, Ncols);
    };
    auto vnlayer = [&](const float* wf, const float* wd,
                       const float* gam, const float* bet,
                       const float* in, long in_bbs, int in_ldb, int Cin,
                       float* pdst, float* ddst, int Cd, float* outdst, long out_bbs,
                       int Cout, int Mlen) {
        const long pbs = (long)Cout * 3 * Mlen;
        const long dbs = (long)Cd * 3 * Mlen;
        gemm(wf, Cin, in, in_bbs, in_ldb, pdst, pbs, (long)3 * Mlen, 1, nullptr, 0, Cout, Cin, 3 * Mlen);
        gemm(wd, Cin, in, in_bbs, in_ldb, ddst, dbs, (long)3 * Mlen, 1, nullptr, 0, Cd, Cin, 3 * Mlen);
        hipMemsetAsync(stats, 0, 2 * Cout * sizeof(float), stream);
        dim3 gs((Mlen + 1023) / 1024, Cout, B);
        vn_norm_stats_kernel<<<gs, 256, 0, stream>>>(pdst, pbs, nbuf, stats, Cout, Mlen);
        const long tot = (long)B * Cout * Mlen;
        vn_finalize_kernel<<<(tot + 255) / 256, 256, 0, stream>>>(
            pdst, pbs, ddst, dbs, Cd, nbuf, stats, gam, bet,
            1.f / ((float)B * (float)Mlen), outdst, out_bbs, Cout, Mlen);
    };
    auto bn1d = [&](float* xb, int Cc, const float* gam, const float* bet, int leaky) {
        hipMemsetAsync(stats, 0, 2 * Cc * sizeof(float), stream);
        dim3 gs(4, Cc, B);
        bn_stats_kernel<<<gs, 256, 0, stream>>>(xb, Cc, stats);
        const long tot = (long)B * Cc * N;
        bn_apply_kernel<<<(tot + 255) / 256, 256, 0, stream>>>(
            xb, Cc, stats, gam, bet, 1.f / ((float)B * (float)N), leaky);
    };

    const long M1 = NK;              // edge-set point count
    const int ld_edge = (int)(3 * NK);

    // ---- Stage 1: knn(x, d=3) + graph feature + VN layers 1,2 ----
    knn_kernel<<<dim3(N / 4, B), 128, 0, stream>>>(x, 3, (long)3 * N, idx);
    {
        const long tot = (long)B * 1 * 3 * N * KNN_K;
        graph_feat_kernel<<<(tot + 255) / 256, 256, 0, stream>>>(x, (long)3 * N, idx, gf, 1);
    }
    vnlayer(w1f, w1d, g1, b1, gf, (long)2 * 3 * NK, ld_edge, 2, pA, dbuf, C, pA, (long)C * 3 * M1, C, (int)M1);
    vnlayer(w2f, w2d, g2, b2, pA, (long)C * 3 * NK, ld_edge, C, pB, dbuf, C, pB, (long)C * 3 * M1, C, (int)M1);
    {
        const long tot = (long)B * C * 3 * N;
        mean_k_kernel<<<(tot + 255) / 256, 256, 0, stream>>>(pB, x123, 0);
    }

    // ---- Stage 2: knn(x1, d=63) + layers 3,4 ----
    knn_kernel<<<dim3(N / 4, B), 128, 0, stream>>>(x123, 63, (long)189 * N, idx);
    {
        const long tot = (long)B * C * 3 * N * KNN_K;
        graph_feat_kernel<<<(tot + 255) / 256, 256, 0, stream>>>(x123, (long)189 * N, idx, gf, C);
    }
    vnlayer(w3f, w3d, g3, b3, gf, (long)42 * 3 * NK, ld_edge, 42, pA, dbuf, C, pA, (long)C * 3 * M1, C, (int)M1);
    vnlayer(w4f, w4d, g4, b4, pA, (long)C * 3 * NK, ld_edge, C, pB, dbuf, C, pB, (long)C * 3 * M1, C, (int)M1);
    {
        const long tot = (long)B * C * 3 * N;
        mean_k_kernel<<<(tot + 255) / 256, 256, 0, stream>>>(pB, x123, 21);
    }

    // ---- Stage 3: knn(x2, d=63) + layer 5 ----
    knn_kernel<<<dim3(N / 4, B), 128, 0, stream>>>(x123 + (long)63 * N, 63, (long)189 * N, idx);
    {
        const long tot = (long)B * C * 3 * N * KNN_K;
        graph_feat_kernel<<<(tot + 255) / 256, 256, 0, stream>>>(x123 + (long)63 * N, (long)189 * N, idx, gf, C);
    }
    vnlayer(w5f, w5d, g5, b5, gf, (long)42 * 3 * NK, ld_edge, 42, pA, dbuf, C, pA, (long)C * 3 * M1, C, (int)M1);
    {
        const long tot = (long)B * C * 3 * N;
        mean_k_kernel<<<(tot + 255) / 256, 256, 0, stream>>>(pA, x123, 42);
    }

    // ---- Layer 6: (341x63) over x123, write into h6cat channels [0,341) ----
    vnlayer(w6f, w6d, g6, b6, x123, (long)189 * N, 3 * N, 63,
            pA, dbuf, 1, h6cat, (long)682 * 3 * N, H, N);
    hmean_cat_kernel<<<dim3(1023, B), 256, 0, stream>>>(h6cat);

    // ---- Std-feature branch: s1 (341x682), s2 (170x341), slin (3x170) ----
    vnlayer(s1f, s1d, sg1, sb1, h6cat, (long)682 * 3 * N, 3 * N, 682,
            pA, dbuf, H, pA, (long)H * 3 * N, H, N);
    vnlayer(s2f, s2d, sg2, sb2, pA, (long)H * 3 * N, 3 * N, H,
            pB, dbuf, H2, pB, (long)H2 * 3 * N, H2, N);
    gemm(slin, H2, pB, (long)H2 * 3 * N, 3 * N, z0, (long)9 * N, (long)3 * N, 1,
         nullptr, 0, 3, H2, 3 * N);

    // ---- hs max (fused) + x123p + w8 bias fold ----
    hsmax_kernel<<<dim3(682, B), 256, 0, stream>>>(h6cat, z0, hmax);
    {
        const long tot = (long)B * 63 * N;
        x123p_kernel<<<(tot + 255) / 256, 256, 0, stream>>>(x123, z0, xp);
    }
    bias8_kernel<<<dim3(256, B), 256, 0, stream>>>(w8, hmax, bias8);

    // ---- MLP head ----
    gemm(w8 + 2046, 2235, xp, (long)189 * N, N, h8, (long)256 * N, (long)N, 1,
         bias8, 256, 256, 189, N);
    bn1d(h8, 256, g8, b8, 1);
    gemm(w9, 256, h8, (long)256 * N, N, h9, (long)256 * N, (long)N, 1, nullptr, 0, 256, 256, N);
    bn1d(h9, 256, g9, b9, 1);
    gemm(w10, 256, h9, (long)256 * N, N, h10, (long)128 * N, (long)N, 1, nullptr, 0, 128, 256, N);
    bn1d(h10, 128, g10, b10, 1);
    // final: write transposed (B, N, 20)
    gemm(w11, 128, h10, (long)128 * N, N, outp, (long)N * 20, 1L, 20, nullptr, 0, 20, 128, N);
}